// MoELayer_36146444763720
// MI455X (gfx1250) — compile-verified
//
#include <hip/hip_runtime.h>
#include <hip/hip_bf16.h>

#define B_DIM 2048
#define H_DIM 768
#define E_DIM 16
#define MAXT  (B_DIM / 16)   // max 16-row tiles per expert
#define BSTRIDE 272          // padded LDS row stride (floats): K-row pairs land
                             // 32 banks apart -> conflict-free ds_load_b32

typedef __attribute__((ext_vector_type(2))) float v2f;
typedef __attribute__((ext_vector_type(8))) float v8f;

__device__ __forceinline__ v8f wmma4f32(v2f a, v2f b, v8f c) {
  // V_WMMA_F32_16X16X4_F32: D(16x16,f32) = A(16x4,f32) x B(4x16,f32) + C
  return __builtin_amdgcn_wmma_f32_16x16x4_f32(false, a, false, b, (short)0, c,
                                               false, false);
}

// Async global->LDS copy of a 16-byte chunk (CDNA5 ASYNCcnt path, no VGPR
// round-trip).  GV addressing: vdst = LDS byte address, vaddr = 64-bit global.
__device__ __forceinline__ void async_cp16(const float* gsrc, const float* ldst) {
  asm volatile("global_load_async_to_lds_b128 %0, %1, off"
               :
               : "v"((unsigned)(unsigned long long)ldst),
                 "v"((unsigned long long)gsrc)
               : "memory");
}
__device__ __forceinline__ void wait_async_le4() {
  asm volatile("s_wait_asynccnt 0x4" ::: "memory");
}
__device__ __forceinline__ void wait_async_0() {
  asm volatile("s_wait_asynccnt 0x0" ::: "memory");
}

// Stage an 8(K) x 256(N) tile of row-major W into LDS (padded rows).
// 128 threads x 4 async B128 chunks = 8*256 floats.
__device__ __forceinline__ void stage_b(const float* __restrict__ Wk,
                                        float (*Bb)[BSTRIDE], int tid) {
#pragma unroll
  for (int q = 0; q < 4; ++q) {
    int chunk = tid + q * 128;        // 0..511
    int kk = chunk >> 6;              // 0..7   (K row)
    int cc = (chunk & 63) << 2;       // 0..252 (col, 4-float chunk)
    async_cp16(Wk + (size_t)kk * H_DIM + cc, &Bb[kk][cc]);
  }
}

// ---------------------------------------------------------------------------
// Kernel 0: zero the per-expert assignment counters (ws state must be reset
// every call for determinism).
// ---------------------------------------------------------------------------
__global__ void zero_counters(int* __restrict__ c1, int* __restrict__ c2) {
  int i = threadIdx.x;
  if (i < E_DIM) { c1[i] = 0; c2[i] = 0; }
}

// ---------------------------------------------------------------------------
// Kernel 1: h = x @ W_lin + b_lin   (2048x768 @ 768x768, fp32 WMMA)
// Block = 4 waves = 32 rows x 256 cols.  B tile double-buffered in LDS via
// async copies; A fed directly (contiguous b64 per lane).  Grid (B/32, H/256).
// ---------------------------------------------------------------------------
__global__ __launch_bounds__(128) void gemm1_kernel(
    const float* __restrict__ x, const float* __restrict__ W,
    const float* __restrict__ bias, float* __restrict__ h) {
  __shared__ float Bbuf[2][8][BSTRIDE];
  const int tid = threadIdx.x;
  const int lane = tid & 31;
  const int wave = tid >> 5;
  const int row0 = blockIdx.x * 32;
  const int colBase = blockIdx.y * 256;
  const int cw = wave * 64;          // this wave's 64-col slice in the tile
  const int lr = lane & 15;          // M (A) / N (B, C)
  const int hk = (lane >> 4) * 2;    // K sub-pair selector (0 or 2)

  v8f acc[2][4];
#pragma unroll
  for (int s = 0; s < 2; ++s)
#pragma unroll
    for (int j = 0; j < 4; ++j) acc[s][j] = (v8f)(0.0f);

  const float* a0p = x + (size_t)(row0 + lr) * H_DIM;
  const float* a1p = x + (size_t)(row0 + 16 + lr) * H_DIM;

  stage_b(W + colBase, Bbuf[0], tid);            // prologue: stage k=0..7
  const int NS = H_DIM / 8;                      // 96 stages
  for (int ks = 0; ks < NS; ++ks) {
    const int buf = ks & 1;
    if (ks + 1 < NS) {
      stage_b(W + (size_t)(ks + 1) * 8 * H_DIM + colBase, Bbuf[buf ^ 1], tid);
      wait_async_le4();                          // oldest 4 (current buf) done
    } else {
      wait_async_0();
    }
    __syncthreads();                             // all waves' staging visible

    const int k0 = ks * 8;
#pragma unroll
    for (int kp = 0; kp < 2; ++kp) {
      v2f a0 = *(const v2f*)(a0p + k0 + kp * 4 + hk);
      v2f a1 = *(const v2f*)(a1p + k0 + kp * 4 + hk);
      const int kb = kp * 4 + hk;
#pragma unroll
      for (int j = 0; j < 4; ++j) {
        v2f b;
        b.x = Bbuf[buf][kb][cw + j * 16 + lr];
        b.y = Bbuf[buf][kb + 1][cw + j * 16 + lr];
        acc[0][j] = wmma4f32(a0, b, acc[0][j]);
        acc[1][j] = wmma4f32(a1, b, acc[1][j]);
      }
    }
    __syncthreads();                             // buf consumed by all waves
  }

  const int mh = (lane >> 4) * 8;  // C/D: high half-lanes hold M+8
#pragma unroll
  for (int s = 0; s < 2; ++s)
#pragma unroll
    for (int j = 0; j < 4; ++j) {
      int c = colBase + cw + j * 16 + lr;
      float bv = bias[c];
#pragma unroll
      for (int i = 0; i < 8; ++i)
        h[(size_t)(row0 + s * 16 + mh + i) * H_DIM + c] = acc[s][j][i] + bv;
    }
}

// ---------------------------------------------------------------------------
// Kernel 2: gating.  One wave per row.  Computes softmax(h_row @ w_gate),
// top-2 experts, normalized gates, keep2 policy, and appends the row to the
// per-expert dispatch lists (integer atomics: slot order is irrelevant to the
// final values -> deterministic output).
// ---------------------------------------------------------------------------
__global__ __launch_bounds__(256) void gating_kernel(
    const float* __restrict__ h, const float* __restrict__ wg,
    const float* __restrict__ noise, float* __restrict__ g1o,
    float* __restrict__ g2o, float* __restrict__ praw,
    int* __restrict__ list1, int* __restrict__ list2,
    int* __restrict__ cnt1, int* __restrict__ cnt2) {
  __shared__ float part[8][32][E_DIM];
  __shared__ float logits[8][E_DIM];
  const int wave = threadIdx.x >> 5;
  const int lane = threadIdx.x & 31;
  const int row = blockIdx.x * 8 + wave;

  float acc[E_DIM];
#pragma unroll
  for (int e = 0; e < E_DIM; ++e) acc[e] = 0.0f;

  for (int i = 0; i < H_DIM / 32; ++i) {
    int k = i * 32 + lane;
    float hv = h[(size_t)row * H_DIM + k];
#pragma unroll
    for (int e = 0; e < E_DIM; ++e) acc[e] += hv * wg[k * E_DIM + e];
  }
#pragma unroll
  for (int e = 0; e < E_DIM; ++e) part[wave][lane][e] = acc[e];
  __syncthreads();

  if (lane < E_DIM) {
    float s = 0.0f;
    for (int l = 0; l < 32; ++l) s += part[wave][l][lane];
    logits[wave][lane] = s;
  }
  __syncthreads();

  if (lane == 0) {
    float l[E_DIM];
    float m = -1e30f;
#pragma unroll
    for (int e = 0; e < E_DIM; ++e) { l[e] = logits[wave][e]; m = fmaxf(m, l[e]); }
    float s = 0.0f;
#pragma unroll
    for (int e = 0; e < E_DIM; ++e) { l[e] = __expf(l[e] - m); s += l[e]; }
    float inv = 1.0f / s;
    int i1 = 0; float g1 = -1.0f;
#pragma unroll
    for (int e = 0; e < E_DIM; ++e) {
      float r = l[e] * inv; l[e] = r;
      if (r > g1) { g1 = r; i1 = e; }
    }
    int i2 = 0; float g2 = -1.0f;
#pragma unroll
    for (int e = 0; e < E_DIM; ++e)
      if (e != i1 && l[e] > g2) { g2 = l[e]; i2 = e; }

    float denom = g1 + g2 + 1e-9f;
    float g1n = g1 / denom;
    float g2n = g2 / denom;
    int keep2 = noise[row] < (g2n / 0.2f);   // THRESHOLD = 0.2

    praw[row] = g1;                 // un-normalized top1 prob (for aux loss)
    g1o[row] = g1n;
    g2o[row] = keep2 ? g2n : 0.0f;

    int s1 = atomicAdd(&cnt1[i1], 1);
    list1[i1 * B_DIM + s1] = row;
    if (keep2) {
      int s2 = atomicAdd(&cnt2[i2], 1);
      list2[i2 * B_DIM + s2] = row;
    }
  }
}

// ---------------------------------------------------------------------------
// Kernel 3: expert GEMM.  Rows grouped per expert into 16-row tiles so the
// B operand (w1[e], 768x768) is wave-uniform; same async-LDS double-buffered
// B feed as gemm1.  Wave computes 16x64 of the expert output, applies ReLU
// and the gate, and writes
//   pass A (accumulate=0): y = h + g1*relu(h @ w1[i1])   (each row once)
//   pass B (accumulate=1): y += g2*relu(h @ w1[i2])       (each row <= once)
// Grid = (E*MAXT, H/256), block = 4 waves.  Padded lanes read row 0 for A
// (WMMA D-rows are independent) and skip the store.
// ---------------------------------------------------------------------------
__global__ __launch_bounds__(128) void expert_kernel(
    const float* __restrict__ h, const float* __restrict__ w1,
    const int* __restrict__ list, const int* __restrict__ cnt,
    const float* __restrict__ gates, float* __restrict__ y, int accumulate) {
  const int e = blockIdx.x / MAXT;
  const int t = blockIdx.x % MAXT;
  const int n = cnt[e];
  if (t * 16 >= n) return;

  __shared__ float Bbuf[2][8][BSTRIDE];
  __shared__ int rows_s[16];
  __shared__ float gate_s[16];
  const int tid = threadIdx.x;
  if (tid < 16) {
    int idx = t * 16 + tid;
    int r = (idx < n) ? list[e * B_DIM + idx] : -1;
    rows_s[tid] = r;
    gate_s[tid] = (r >= 0) ? gates[r] : 0.0f;
  }
  __syncthreads();

  const int lane = tid & 31;
  const int wave = tid >> 5;
  const int colBase = blockIdx.y * 256;
  const int cw = wave * 64;
  const int lr = lane & 15;
  const int hk = (lane >> 4) * 2;

  const int arow = rows_s[lr];
  const float* aptr = h + (size_t)(arow < 0 ? 0 : arow) * H_DIM;
  const float* W = w1 + (size_t)e * H_DIM * H_DIM;

  v8f acc[4];
#pragma unroll
  for (int j = 0; j < 4; ++j) acc[j] = (v8f)(0.0f);

  stage_b(W + colBase, Bbuf[0], tid);
  const int NS = H_DIM / 8;
  for (int ks = 0; ks < NS; ++ks) {
    const int buf = ks & 1;
    if (ks + 1 < NS) {
      stage_b(W + (size_t)(ks + 1) * 8 * H_DIM + colBase, Bbuf[buf ^ 1], tid);
      wait_async_le4();
    } else {
      wait_async_0();
    }
    __syncthreads();

    const int k0 = ks * 8;
#pragma unroll
    for (int kp = 0; kp < 2; ++kp) {
      v2f a = *(const v2f*)(aptr + k0 + kp * 4 + hk);
      const int kb = kp * 4 + hk;
#pragma unroll
      for (int j = 0; j < 4; ++j) {
        v2f b;
        b.x = Bbuf[buf][kb][cw + j * 16 + lr];
        b.y = Bbuf[buf][kb + 1][cw + j * 16 + lr];
        acc[j] = wmma4f32(a, b, acc[j]);
      }
    }
    __syncthreads();
  }

  const int mh = (lane >> 4) * 8;
#pragma unroll
  for (int j = 0; j < 4; ++j) {
    int c = colBase + cw + j * 16 + lr;
#pragma unroll
    for (int i = 0; i < 8; ++i) {
      int m = mh + i;
      int r = rows_s[m];
      if (r < 0) continue;
      float v = acc[j][i];
      v = v > 0.0f ? v : 0.0f;
      v *= gate_s[m];
      size_t off = (size_t)r * H_DIM + c;
      if (accumulate) y[off] += v;
      else            y[off] = h[off] + v;
    }
  }
}

// ---------------------------------------------------------------------------
// Kernel 4: BatchNorm batch statistics per column (biased variance, as
// jnp.var).  One thread per column, coalesced row sweeps.
// ---------------------------------------------------------------------------
__global__ __launch_bounds__(256) void bn_stats(const float* __restrict__ y,
                                                float* __restrict__ mean,
                                                float* __restrict__ rstd) {
  int c = blockIdx.x * blockDim.x + threadIdx.x;
  if (c >= H_DIM) return;
  float s = 0.0f, s2 = 0.0f;
  for (int r = 0; r < B_DIM; ++r) {
    float v = y[(size_t)r * H_DIM + c];
    s += v; s2 += v * v;
  }
  float m = s * (1.0f / B_DIM);
  float var = s2 * (1.0f / B_DIM) - m * m;
  mean[c] = m;
  rstd[c] = rsqrtf(var + 1e-5f);
}

// ---------------------------------------------------------------------------
// Kernel 5: normalize + affine + ReLU -> d_out
// ---------------------------------------------------------------------------
__global__ __launch_bounds__(256) void bn_apply(
    const float* __restrict__ y, const float* __restrict__ mean,
    const float* __restrict__ rstd, const float* __restrict__ gamma,
    const float* __restrict__ beta, float* __restrict__ out) {
  int i = blockIdx.x * blockDim.x + threadIdx.x;
  if (i >= B_DIM * H_DIM) return;
  int c = i % H_DIM;
  float v = (y[i] - mean[c]) * rstd[c] * gamma[c] + beta[c];
  out[i] = v > 0.0f ? v : 0.0f;
}

// ---------------------------------------------------------------------------
// Kernel 6: deterministic tree reduction for the aux loss.
// loss = sum_b(raw[argmax]) * (E/B) * LOSS_COEF
// ---------------------------------------------------------------------------
__global__ __launch_bounds__(256) void loss_kernel(const float* __restrict__ praw,
                                                   float* __restrict__ out_loss) {
  __shared__ float red[256];
  float s = 0.0f;
  for (int i = threadIdx.x; i < B_DIM; i += 256) s += praw[i];
  red[threadIdx.x] = s;
  __syncthreads();
  for (int off = 128; off > 0; off >>= 1) {
    if ((int)threadIdx.x < off) red[threadIdx.x] += red[threadIdx.x + off];
    __syncthreads();
  }
  if (threadIdx.x == 0)
    out_loss[0] = red[0] * ((float)E_DIM / (float)B_DIM) * 1e-2f;
}

// ---------------------------------------------------------------------------
extern "C" void kernel_launch(void* const* d_in, const int* in_sizes, int n_in,
                              void* d_out, int out_size, void* d_ws, size_t ws_size,
                              hipStream_t stream) {
  (void)in_sizes; (void)n_in; (void)out_size; (void)ws_size;

  const float* x      = (const float*)d_in[0];  // [B, H]
  const float* noise  = (const float*)d_in[1];  // [B, 1]
  const float* W_lin  = (const float*)d_in[2];  // [H, H]
  const float* b_lin  = (const float*)d_in[3];  // [H]
  const float* w_gate = (const float*)d_in[4];  // [H, E]
  const float* w1     = (const float*)d_in[5];  // [E, H, H]
  const float* gamma  = (const float*)d_in[6];  // [H]
  const float* beta   = (const float*)d_in[7];  // [H]
  float* out = (float*)d_out;                   // [B*H] y_relu, then [1] loss

  // Workspace layout
  char* ws = (char*)d_ws;
  float* h    = (float*)ws;                                     // B*H
  float* y    = (float*)(ws + (size_t)B_DIM * H_DIM * 4);       // B*H
  float* g1o  = (float*)(ws + 2ull * B_DIM * H_DIM * 4);        // B
  float* g2o  = g1o + B_DIM;                                    // B
  float* praw = g2o + B_DIM;                                    // B
  float* mean = praw + B_DIM;                                   // H
  float* rstd = mean + H_DIM;                                   // H
  int*   cnt1 = (int*)(rstd + H_DIM);                           // E
  int*   cnt2 = cnt1 + E_DIM;                                   // E
  int*   list1 = cnt2 + E_DIM;                                  // E*B
  int*   list2 = list1 + E_DIM * B_DIM;                         // E*B

  zero_counters<<<1, 64, 0, stream>>>(cnt1, cnt2);

  gemm1_kernel<<<dim3(B_DIM / 32, H_DIM / 256), 128, 0, stream>>>(x, W_lin,
                                                                  b_lin, h);

  gating_kernel<<<B_DIM / 8, 256, 0, stream>>>(h, w_gate, noise, g1o, g2o,
                                               praw, list1, list2, cnt1, cnt2);

  expert_kernel<<<dim3(E_DIM * MAXT, H_DIM / 256), 128, 0, stream>>>(
      h, w1, list1, cnt1, g1o, y, 0);
  expert_kernel<<<dim3(E_DIM * MAXT, H_DIM / 256), 128, 0, stream>>>(
      h, w1, list2, cnt2, g2o, y, 1);

  bn_stats<<<(H_DIM + 255) / 256, 256, 0, stream>>>(y, mean, rstd);
  bn_apply<<<((size_t)B_DIM * H_DIM + 255) / 256, 256, 0, stream>>>(
      y, mean, rstd, gamma, beta, out);

  loss_kernel<<<1, 256, 0, stream>>>(praw, out + (size_t)B_DIM * H_DIM);
}